// Model_30365418783206
// MI455X (gfx1250) — compile-verified
//
#include <hip/hip_runtime.h>

// Problem constants (fixed by the reference).
#define BB 128
#define LL 512
#define NN 321
#define KERN 25
#define PAD 12   // (KERN-1)/2

typedef __attribute__((ext_vector_type(2))) float v2f;
typedef __attribute__((ext_vector_type(8))) float v8f;

// ---------------------------------------------------------------------------
// Kernel 1: column reductions of W via V_WMMA_F32_16X16X4_F32.
//   w0[l] = sum_p W[p,l]
//   w1[l] = sum_p (p - L) * W[p,l]
// One 1024-thread block = 32 waves; wave w owns l-tile [16w, 16w+16).
// A (16x4 f32, 2 VGPRs): lane m=lane&15; lane<16 -> K={k0,k0+1}, lane>=16 ->
// K={k0+2,k0+3}. Row M=0 is all-ones, row M=1 is t_old[K], rows 2..15 zero.
// B (4x16 f32, 2 VGPRs): mirrored layout, element = W[K, l0+m].
// C/D (16x16 f32, 8 VGPRs): lanes 0-15, vgpr r -> (M=r, N=lane).
// ---------------------------------------------------------------------------
__global__ void colsum_wmma_kernel(const float* __restrict__ W,
                                   float* __restrict__ w0,
                                   float* __restrict__ w1) {
    const int wave = threadIdx.x >> 5;   // 0..31
    const int lane = threadIdx.x & 31;
    const int m    = lane & 15;
    const int hi   = (lane >= 16) ? 1 : 0;
    const int l0   = wave * 16;

    v8f acc = {};
    for (int k0 = 0; k0 < LL; k0 += 4) {
        const int ka = k0 + (hi ? 2 : 0);   // K for element .x of this lane
        const int kb = ka + 1;              // K for element .y

        v2f a;
        a.x = (m == 0) ? 1.0f : ((m == 1) ? (float)(ka - LL) : 0.0f);
        a.y = (m == 0) ? 1.0f : ((m == 1) ? (float)(kb - LL) : 0.0f);

        v2f bmat;
        bmat.x = W[(size_t)ka * LL + l0 + m];
        bmat.y = W[(size_t)kb * LL + l0 + m];

        acc = __builtin_amdgcn_wmma_f32_16x16x4_f32(
            /*neg_a=*/false, a, /*neg_b=*/false, bmat,
            /*c_mod=*/(short)0, acc, /*reuse_a=*/false, /*reuse_b=*/false);
    }

    if (!hi) {              // lanes 0-15: vgpr0 = row M=0, vgpr1 = row M=1
        w0[l0 + m] = acc[0];
        w1[l0 + m] = acc[1];
    }
}

// ---------------------------------------------------------------------------
// Kernel 2: fold the moving-average adjoint into the weights and compute the
// bias constants.
//   hw[j] = (1/K) * sum_l count(j,l) * w[l]
// where count(j,l) = multiplicity with which x[j] enters trend[l] under
// replicate padding (interior: 1 iff |j-l|<=pad; j=0 / j=L-1 get the
// clamped-end multiplicities).
//   cvec = { sum_p b[p], sum_p (p-L)*b[p] }
// One 512-thread block.
// ---------------------------------------------------------------------------
__global__ void adjoint_kernel(const float* __restrict__ w0,
                               const float* __restrict__ w1,
                               const float* __restrict__ bt,
                               float* __restrict__ hw0,
                               float* __restrict__ hw1,
                               float* __restrict__ cvec) {
    const int j = threadIdx.x;   // 0..511
    float a0 = 0.0f, a1 = 0.0f;
    for (int l = 0; l < LL; ++l) {
        float cnt;
        if (j == 0) {
            cnt = (l <= PAD) ? (float)(PAD - l + 1) : 0.0f;
        } else if (j == LL - 1) {
            cnt = (l >= LL - 1 - PAD) ? (float)(l + PAD - (LL - 1) + 1) : 0.0f;
        } else {
            cnt = (l - PAD <= j && j <= l + PAD) ? 1.0f : 0.0f;
        }
        a0 = fmaf(cnt, w0[l], a0);
        a1 = fmaf(cnt, w1[l], a1);
    }
    const float rk = 1.0f / (float)KERN;
    hw0[j] = a0 * rk;
    hw1[j] = a1 * rk;

    if (j == 0) {
        float c0 = 0.0f, c1 = 0.0f;
        for (int p = 0; p < LL; ++p) {
            c0 += bt[p];
            c1 = fmaf((float)(p - LL), bt[p], c1);
        }
        cvec[0] = c0;
        cvec[1] = c1;
    }
}

// ---------------------------------------------------------------------------
// Kernel 3: bandwidth-bound main pass. One thread per (b, n).
//   s0 = c0 + sum_l x[b,l,n]*hw0[l];  s1 = c1 + sum_l x[b,l,n]*hw1[l]
//   theta = [s0,s1] * inv(XTX)  (constant 2x2)
//   out[b,p,n] = theta0 + p*theta1
// Coalesced over n (stride-1); weights broadcast from LDS.
// ---------------------------------------------------------------------------
__global__ void predict_kernel(const float* __restrict__ x,
                               const float* __restrict__ hw0,
                               const float* __restrict__ hw1,
                               const float* __restrict__ cvec,
                               float* __restrict__ out) {
    __shared__ float sw0[LL];
    __shared__ float sw1[LL];
    for (int i = threadIdx.x; i < LL; i += blockDim.x) {
        sw0[i] = hw0[i];
        sw1[i] = hw1[i];
    }
    __syncthreads();

    const int n = blockIdx.x * blockDim.x + threadIdx.x;
    const int b = blockIdx.y;
    if (n >= NN) return;

    const float* xb = x + (size_t)b * LL * NN + n;
    float s0 = cvec[0];
    float s1 = cvec[1];
#pragma unroll 8
    for (int l = 0; l < LL; ++l) {
        const float v = xb[(size_t)l * NN];
        s0 = fmaf(v, sw0[l], s0);
        s1 = fmaf(v, sw1[l], s1);
    }

    // inv(XTX) for X = [1, t], t in [-512, -1]:
    //   St  = -131328, St2 = 44870400, det = 512*St2 - St^2 = 5726601216
    const float St      = -131328.0f;
    const float St2     = 44870400.0f;
    const float inv_det = 1.0f / 5726601216.0f;
    const float th0 = (s0 * St2 - s1 * St) * inv_det;
    const float th1 = (s1 * (float)LL - s0 * St) * inv_det;

    float* ob = out + (size_t)b * LL * NN + n;
#pragma unroll 8
    for (int p = 0; p < LL; ++p) {
        ob[(size_t)p * NN] = fmaf((float)p, th1, th0);
    }
}

extern "C" void kernel_launch(void* const* d_in, const int* in_sizes, int n_in,
                              void* d_out, int out_size, void* d_ws, size_t ws_size,
                              hipStream_t stream) {
    const float* x_enc   = (const float*)d_in[0];   // [B, L, N]
    const float* W_trend = (const float*)d_in[1];   // [L, L]
    const float* b_trend = (const float*)d_in[2];   // [L]
    float* out = (float*)d_out;                     // [B, L, N]

    float* ws   = (float*)d_ws;
    float* w0   = ws;            // [512] raw column sums
    float* w1   = ws + 512;      // [512]
    float* hw0  = ws + 1024;     // [512] after moving-avg adjoint
    float* hw1  = ws + 1536;     // [512]
    float* cvec = ws + 2048;     // [2]  bias constants

    // 1) WMMA column reductions of W (32 waves, one block).
    colsum_wmma_kernel<<<1, 1024, 0, stream>>>(W_trend, w0, w1);

    // 2) Fold moving-average adjoint + bias constants.
    adjoint_kernel<<<1, LL, 0, stream>>>(w0, w1, b_trend, hw0, hw1, cvec);

    // 3) Streaming main pass: grid over (ceil(N/128), B).
    dim3 grid((NN + 127) / 128, BB);
    predict_kernel<<<grid, 128, 0, stream>>>(x_enc, hw0, hw1, cvec, out);
}